// HyperKA_28123445854707
// MI455X (gfx1250) — compile-verified
//
#include <hip/hip_runtime.h>
#include <math.h>

#define D 128
#define MIN_NORM 1e-15f
#define PROJ_EPS 1e-5f
#define CRW 0.1f   // COMBINE_RELS_WEIGHT

typedef __attribute__((ext_vector_type(2))) float v2f;
typedef __attribute__((ext_vector_type(8))) float v8f;

#if defined(__HIP_DEVICE_COMPILE__) && __has_builtin(__builtin_amdgcn_wmma_f32_16x16x4_f32)
#define HAVE_WMMA_F32 1
#else
#define HAVE_WMMA_F32 0
#endif

// ---------- helpers ----------

__device__ __forceinline__ float wsum(float v) {
  // full butterfly: result valid in all 32 lanes (wave32)
  #pragma unroll
  for (int off = 16; off > 0; off >>= 1) v += __shfl_xor(v, off, 32);
  return v;
}

__device__ __forceinline__ float rdot(const float* a, const float* b) {
  return wsum(a[0] * b[0] + a[1] * b[1] + a[2] * b[2] + a[3] * b[3]);
}

// order-preserving float <-> int mapping so atomicMax(int*) == float max
__device__ __forceinline__ int f2oi(float f) {
  int i = __float_as_int(f);
  return (i >= 0) ? i : (i ^ 0x7FFFFFFF);
}
__device__ __forceinline__ float oi2f(int i) {
  return __int_as_float((i >= 0) ? i : (i ^ 0x7FFFFFFF));
}

// ---------- 1. t = log_map_zero(ents), wave per row ----------

__global__ void k_logmap(const float* __restrict__ ents, float* __restrict__ t, int nrows) {
  int row = blockIdx.x * 8 + (threadIdx.x >> 5);
  int lane = threadIdx.x & 31;
  if (row >= nrows) return;
  float4 x = reinterpret_cast<const float4*>(ents + (size_t)row * D)[lane];
  float ss = wsum(x.x * x.x + x.y * x.y + x.z * x.z + x.w * x.w);
  float n = sqrtf(fmaxf(ss, MIN_NORM));
  float nc = fminf(fmaxf(n, MIN_NORM), 1.0f - PROJ_EPS);
  float s = atanhf(nc) / nc;
  float4 o = make_float4(x.x * s, x.y * s, x.z * s, x.w * s);
  reinterpret_cast<float4*>(t + (size_t)row * D)[lane] = o;
}

// ---------- 1b. WT[n*D + k] = W[k*D + n]  (one-off 64 KB transpose) ----------

__global__ void k_transposeW(const float* __restrict__ W, float* __restrict__ WT) {
  int idx = blockIdx.x * 256 + threadIdx.x;   // idx over D*D
  if (idx >= D * D) return;
  int n = idx / D;
  int k = idx - n * D;
  WT[idx] = W[k * D + n];
}

// ---------- 2. m = t @ W  (fp32 WMMA, 16x16x4) ----------
// block = 256 threads = 8 waves; block computes rows [16*bid, 16*bid+16) x all 128 cols.
// wave w owns N-tile [16w, 16w+16). A tile staged in LDS; B fragments are
// contiguous 64-bit loads from the pre-transposed WT (L2-resident, 64 KB).

__global__ void k_gemm(const float* __restrict__ T, const float* __restrict__ WT,
                       float* __restrict__ M, int nrows) {
  __shared__ float sA[16 * D];  // 8 KB
  int row0 = blockIdx.x * 16;
  int tid = threadIdx.x;
  for (int i = tid; i < 16 * (D / 4); i += 256) {
    int r = i / (D / 4);
    int cc = i - r * (D / 4);
    int gr = row0 + r;
    float4 v = make_float4(0.f, 0.f, 0.f, 0.f);
    if (gr < nrows) v = reinterpret_cast<const float4*>(T + (size_t)gr * D)[cc];
    reinterpret_cast<float4*>(sA)[i] = v;
  }
  __syncthreads();

  int lane = tid & 31;
  int n0 = (tid >> 5) * 16;   // wave's N tile
  int laneM = lane & 15;
  int laneHi = lane >> 4;     // 0: lanes 0-15, 1: lanes 16-31

  const float* wrow = WT + (size_t)(n0 + laneM) * D;  // this lane's W column
  const float* arow = sA + laneM * D;

  v8f acc = {};
  #pragma unroll
  for (int k0 = 0; k0 < D; k0 += 4) {
    int ka = k0 + 2 * laneHi;
#if HAVE_WMMA_F32
    // A 16x4: lane -> M=laneM, VGPR j + lane-half -> K = k0 + 2*laneHi + j
    v2f a, b;
    a[0] = arow[ka];
    a[1] = arow[ka + 1];
    // B 4x16: lane -> N=n0+laneM; contiguous pair in WT -> one b64 load
    b = *reinterpret_cast<const v2f*>(wrow + ka);
    acc = __builtin_amdgcn_wmma_f32_16x16x4_f32(false, a, false, b, (short)0, acc,
                                                false, false);
#else
    (void)ka;
    #pragma unroll
    for (int kk = 0; kk < 4; kk++) {
      float bb = wrow[k0 + kk];
      #pragma unroll
      for (int v = 0; v < 8; v++)
        acc[v] += sA[(v + 8 * laneHi) * D + k0 + kk] * bb;
    }
#endif
  }
  // C/D layout: VGPR v: lanes0-15 -> M=v, lanes16-31 -> M=v+8; N = lane%16
  #pragma unroll
  for (int v = 0; v < 8; v++) {
    int gr = row0 + v + 8 * laneHi;
    if (gr < nrows) M[(size_t)gr * D + n0 + laneM] = acc[v];
  }
}

// ---------- 3. init accumulators (runs AFTER gemm; relacc aliases t) ----------

__global__ void k_init(float* __restrict__ outp, float* __restrict__ relacc,
                       float* __restrict__ ssum, int* __restrict__ smax,
                       int n, int nd) {
  int i = blockIdx.x * blockDim.x + threadIdx.x;
  int stride = gridDim.x * blockDim.x;
  for (int j = i; j < nd; j += stride) { outp[j] = 0.f; relacc[j] = 0.f; }
  for (int j = i; j < n; j += stride) { ssum[j] = 0.f; smax[j] = (int)0x807FFFFF; } // f2oi(-inf)
}

// ---------- 4. bias_h = proj(exp_map_zero(bias_vec)), one wave ----------

__global__ void k_bias(const float* __restrict__ bias_vec, float* __restrict__ bias_h) {
  int lane = threadIdx.x & 31;
  float4 v = reinterpret_cast<const float4*>(bias_vec)[lane];
  float x[4] = {v.x, v.y, v.z, v.w};
  float n = sqrtf(fmaxf(rdot(x, x), MIN_NORM));
  float s = tanhf(n) / n;
  #pragma unroll
  for (int j = 0; j < 4; j++) x[j] *= s;
  n = sqrtf(fmaxf(rdot(x, x), MIN_NORM));
  s = fminf(1.0f, (1.0f - PROJ_EPS) / n);
  #pragma unroll
  for (int j = 0; j < 4; j++) x[j] *= s;
  reinterpret_cast<float4*>(bias_h)[lane] = make_float4(x[0], x[1], x[2], x[3]);
}

// ---------- 5. edge scores + segment max, wave per edge ----------

__global__ void k_score(const float* __restrict__ m, const int* __restrict__ er,
                        const int* __restrict__ ec, float* __restrict__ scores,
                        int* __restrict__ smax, int E) {
  int e = blockIdx.x * 8 + (threadIdx.x >> 5);
  int lane = threadIdx.x & 31;
  if (e >= E) return;
  int r = er[e], c = ec[e];
  float4 a = reinterpret_cast<const float4*>(m + (size_t)r * D)[lane];
  float4 b = reinterpret_cast<const float4*>(m + (size_t)c * D)[lane];
  float s = wsum(a.x * b.x + a.y * b.y + a.z * b.z + a.w * b.w);
  if (lane == 0) {
    scores[e] = s;
    atomicMax(&smax[r], f2oi(s));
  }
}

// ---------- 6. ex = exp(score - max); segment sum ----------

__global__ void k_expsum(const int* __restrict__ smax, const int* __restrict__ er,
                         float* __restrict__ scores, float* __restrict__ ssum, int E) {
  int e = blockIdx.x * 256 + threadIdx.x;
  if (e >= E) return;
  int r = er[e];
  float ex = expf(scores[e] - oi2f(smax[r]));
  scores[e] = ex;  // reuse buffer
  atomicAdd(&ssum[r], ex);
}

// ---------- 7. near_ents += alpha * m[col], wave per edge ----------

__global__ void k_agg(const float* __restrict__ m, const int* __restrict__ er,
                      const int* __restrict__ ec, const float* __restrict__ exv,
                      const float* __restrict__ ssum, float* __restrict__ acc, int E) {
  int e = blockIdx.x * 8 + (threadIdx.x >> 5);
  int lane = threadIdx.x & 31;
  if (e >= E) return;
  int r = er[e], c = ec[e];
  float alpha = exv[e] / fmaxf(ssum[r], MIN_NORM);
  float4 b = reinterpret_cast<const float4*>(m + (size_t)c * D)[lane];
  float* dst = acc + (size_t)r * D + lane * 4;
  atomicAdd(dst + 0, alpha * b.x);
  atomicAdd(dst + 1, alpha * b.y);
  atomicAdd(dst + 2, alpha * b.z);
  atomicAdd(dst + 3, alpha * b.w);
}

// ---------- 8. near_rels += rels[val], wave per rel-edge ----------

__global__ void k_relagg(const float* __restrict__ rels, const int* __restrict__ rr,
                         const int* __restrict__ rv, float* __restrict__ acc, int E) {
  int e = blockIdx.x * 8 + (threadIdx.x >> 5);
  int lane = threadIdx.x & 31;
  if (e >= E) return;
  int r = rr[e], v = rv[e];
  float4 b = reinterpret_cast<const float4*>(rels + (size_t)v * D)[lane];
  float* dst = acc + (size_t)r * D + lane * 4;
  atomicAdd(dst + 0, b.x);
  atomicAdd(dst + 1, b.y);
  atomicAdd(dst + 2, b.z);
  atomicAdd(dst + 3, b.w);
}

// ---------- 9. fused hyperbolic tail, wave per row, in place on out ----------

__global__ void k_final(float* __restrict__ out, const float* __restrict__ relacc,
                        const float* __restrict__ cnts, const float* __restrict__ bias_h,
                        int nrows) {
  int row = blockIdx.x * 8 + (threadIdx.x >> 5);
  int lane = threadIdx.x & 31;
  if (row >= nrows) return;
  size_t base = (size_t)row * D + lane * 4;
  float4 ne = *reinterpret_cast<const float4*>(out + base);
  float4 nr = *reinterpret_cast<const float4*>(relacc + base);
  float inv_cnt = 1.0f / cnts[row];
  float x[4] = {ne.x + CRW * nr.x * inv_cnt, ne.y + CRW * nr.y * inv_cnt,
                ne.z + CRW * nr.z * inv_cnt, ne.w + CRW * nr.w * inv_cnt};

  // exp_map_zero
  float n = sqrtf(fmaxf(rdot(x, x), MIN_NORM));
  float s = tanhf(n) / n;
  #pragma unroll
  for (int j = 0; j < 4; j++) x[j] *= s;
  // projection
  n = sqrtf(fmaxf(rdot(x, x), MIN_NORM));
  s = fminf(1.0f, (1.0f - PROJ_EPS) / n);
  #pragma unroll
  for (int j = 0; j < 4; j++) x[j] *= s;

  // mobius_addition(x, bias_h)
  float4 bh = reinterpret_cast<const float4*>(bias_h)[lane];
  float y[4] = {bh.x, bh.y, bh.z, bh.w};
  float x2 = rdot(x, x), y2 = rdot(y, y), xy = rdot(x, y);
  float ca = 1.0f + 2.0f * xy + y2;
  float cb = 1.0f - x2;
  float den = fmaxf(1.0f + 2.0f * xy + x2 * y2, MIN_NORM);
  #pragma unroll
  for (int j = 0; j < 4; j++) x[j] = (ca * x[j] + cb * y[j]) / den;
  // projection
  n = sqrtf(fmaxf(rdot(x, x), MIN_NORM));
  s = fminf(1.0f, (1.0f - PROJ_EPS) / n);
  #pragma unroll
  for (int j = 0; j < 4; j++) x[j] *= s;

  // log_map_zero
  n = sqrtf(fmaxf(rdot(x, x), MIN_NORM));
  float nc = fminf(fmaxf(n, MIN_NORM), 1.0f - PROJ_EPS);
  s = atanhf(nc) / nc;
  #pragma unroll
  for (int j = 0; j < 4; j++) x[j] = tanhf(x[j] * s);   // tanh in tangent space
  // exp_map_zero
  n = sqrtf(fmaxf(rdot(x, x), MIN_NORM));
  s = tanhf(n) / n;
  #pragma unroll
  for (int j = 0; j < 4; j++) x[j] *= s;
  // projection
  n = sqrtf(fmaxf(rdot(x, x), MIN_NORM));
  s = fminf(1.0f, (1.0f - PROJ_EPS) / n);
  #pragma unroll
  for (int j = 0; j < 4; j++) x[j] *= s;

  *reinterpret_cast<float4*>(out + base) = make_float4(x[0], x[1], x[2], x[3]);
}

// ---------- host ----------

extern "C" void kernel_launch(void* const* d_in, const int* in_sizes, int n_in,
                              void* d_out, int out_size, void* d_ws, size_t ws_size,
                              hipStream_t stream) {
  (void)n_in; (void)out_size; (void)ws_size;
  const float* ents = (const float*)d_in[0];
  const float* rels = (const float*)d_in[1];
  const float* W    = (const float*)d_in[2];
  const float* bias = (const float*)d_in[3];
  const float* cnts = (const float*)d_in[4];
  const int* er = (const int*)d_in[5];
  const int* ec = (const int*)d_in[6];
  const int* rr = (const int*)d_in[7];
  const int* rv = (const int*)d_in[8];
  const int N  = in_sizes[0] / D;
  const int E  = in_sizes[5];
  const int ER = in_sizes[7];
  float* out = (float*)d_out;

  char* p = (char*)d_ws;
  auto take = [&](size_t bytes) {
    char* r = p;
    p += (bytes + 255) & ~(size_t)255;
    return (void*)r;
  };
  float* ws_t    = (float*)take((size_t)N * D * sizeof(float)); // t, reused as rel acc
  float* ws_m    = (float*)take((size_t)N * D * sizeof(float));
  float* ws_sc   = (float*)take((size_t)E * sizeof(float));
  float* ws_ssum = (float*)take((size_t)N * sizeof(float));
  int*   ws_smax = (int*)take((size_t)N * sizeof(int));
  float* ws_bias = (float*)take(D * sizeof(float));
  float* ws_wt   = (float*)take((size_t)D * D * sizeof(float)); // W transposed

  const int rowBlocks = (N + 7) / 8;   // wave-per-row kernels
  const int eBlocks   = (E + 7) / 8;   // wave-per-edge kernels
  const int erBlocks  = (ER + 7) / 8;

  k_logmap<<<rowBlocks, 256, 0, stream>>>(ents, ws_t, N);
  k_transposeW<<<(D * D + 255) / 256, 256, 0, stream>>>(W, ws_wt);
  k_gemm<<<(N + 15) / 16, 256, 0, stream>>>(ws_t, ws_wt, ws_m, N);
  k_init<<<2048, 256, 0, stream>>>(out, ws_t, ws_ssum, ws_smax, N, N * D);
  k_bias<<<1, 32, 0, stream>>>(bias, ws_bias);
  k_score<<<eBlocks, 256, 0, stream>>>(ws_m, er, ec, ws_sc, ws_smax, E);
  k_expsum<<<(E + 255) / 256, 256, 0, stream>>>(ws_smax, er, ws_sc, ws_ssum, E);
  k_agg<<<eBlocks, 256, 0, stream>>>(ws_m, er, ec, ws_sc, ws_ssum, out, E);
  k_relagg<<<erBlocks, 256, 0, stream>>>(rels, rr, rv, ws_t, ER);
  k_final<<<rowBlocks, 256, 0, stream>>>(out, ws_t, cnts, ws_bias, N);
}